// CLDS_39298950758798
// MI455X (gfx1250) — compile-verified
//
#include <hip/hip_runtime.h>
#include <hip/hip_bf16.h>
#include <math.h>

typedef __attribute__((ext_vector_type(2))) float v2f;
typedef __attribute__((ext_vector_type(8))) float v8f;

#define NUV 100000
#define NIV 150000
#define NV  (NUV + NIV)
#define DV  64
#define NNZ_A 2000000
#define NNZ_S 800000

// ---------------------------------------------------------------------------
// init: E0 = concat(0.5*(u1+u2), item_emb); acc(d_out) = E0  (embs[0] term)
// ---------------------------------------------------------------------------
__global__ void k_init(const float* __restrict__ u1, const float* __restrict__ u2,
                       const float* __restrict__ items,
                       float* __restrict__ E0, float* __restrict__ acc,
                       size_t nUser, size_t nTot) {
  size_t i = (size_t)blockIdx.x * blockDim.x + threadIdx.x;
  if (i >= nTot) return;
  float v = (i < nUser) ? 0.5f * (u1[i] + u2[i]) : items[i - nUser];
  E0[i] = v;
  acc[i] = v;
}

// n1 = u1[shuffle1], n2 = u2[shuffle2]
__global__ void k_gather(const float* __restrict__ u1, const float* __restrict__ u2,
                         const int* __restrict__ s1, const int* __restrict__ s2,
                         float* __restrict__ n1, float* __restrict__ n2, size_t nUD) {
  size_t i = (size_t)blockIdx.x * blockDim.x + threadIdx.x;
  if (i >= nUD) return;
  size_t r = i >> 6;
  int d = (int)(i & 63);
  n1[i] = u1[((size_t)s1[r] << 6) | d];
  n2[i] = u2[((size_t)s2[r] << 6) | d];
}

// ---------------------------------------------------------------------------
// SpMM scatter-add: out[rows[e]] += vals[e] * X[cols[e]]; 16 threads/edge,
// each thread handles 4 consecutive floats -> float4 load + 4 f32 atomics
// (global_atomic_add_f32 via unsafeAtomicAdd).
// ---------------------------------------------------------------------------
__global__ void k_spmm(const int* __restrict__ rows, const int* __restrict__ cols,
                       const float* __restrict__ vals, const float* __restrict__ X,
                       float* __restrict__ out, long long nnz) {
  long long t = (long long)blockIdx.x * blockDim.x + threadIdx.x;
  long long e = t >> 4;
  if (e >= nnz) return;
  int c4 = (int)(t & 15) * 4;
  int r = rows[e], c = cols[e];
  float v = vals[e];
  float4 x = *(const float4*)(X + ((size_t)c << 6) + c4);
  float* o = out + ((size_t)r << 6) + c4;
  unsafeAtomicAdd(o + 0, v * x.x);
  unsafeAtomicAdd(o + 1, v * x.y);
  unsafeAtomicAdd(o + 2, v * x.z);
  unsafeAtomicAdd(o + 3, v * x.w);
}

// ---------------------------------------------------------------------------
// out(M x 64) = X(M x K) @ W(64 x K)^T, f32 WMMA, compile-time K/AVG/TANH.
// K = 64 (X = X0) or 128 (cols 0..63 from X0, cols 64..127 = 0.5*(X1a+X1b)).
// 256 threads = 8 waves; each wave owns one 16-row stripe (16x64 output).
// Per 16x16 tile: K/4 x V_WMMA_F32_16X16X4_F32 over K. W staged in LDS once
// per block (float4 fill). Wave-uniform m0 guard keeps EXEC all-ones inside
// WMMA. In-place (out==X0) is safe: each wave reads only its own 16 rows and
// all its reads precede its stores.
//
// Fragment layouts (ISA 7.12.2):
//   A 16x4 f32 : lane<16 -> M=lane, K=k+0,k+1 ; lane>=16 -> M=lane-16, K=k+2,k+3
//   B 4x16 f32 : VGPR0 = rows k+0 (lanes 0-15) / k+2 (lanes 16-31), VGPR1 = +1
//   C/D        : VGPR v -> row m0+v (lanes<16) or m0+8+v, col = lane&15
// ---------------------------------------------------------------------------
template <int K, bool AVG, bool TANH>
__global__ void k_gemm_wmma(const float* X0, const float* X1a, const float* X1b,
                            const float* __restrict__ W, float* out, int M) {
  __shared__ float sW[64 * K];
  int tid = threadIdx.x;
  for (int i = tid * 4; i < 64 * K; i += blockDim.x * 4)
    *(float4*)(sW + i) = *(const float4*)(W + i);
  __syncthreads();

  int wave = tid >> 5;
  int lane = tid & 31;
  int m0 = (blockIdx.x * 8 + wave) * 16;
  if (m0 >= M) return;  // wave-uniform: EXEC stays all-ones for WMMA

  int mrow = m0 + (lane & 15);
  int khalf = (lane >> 4) << 1;  // 0 for lanes 0-15, 2 for lanes 16-31
  int nl = lane & 15;

  v8f acc0 = {}, acc1 = {}, acc2 = {}, acc3 = {};

#pragma unroll
  for (int k = 0; k < K; k += 4) {
    int kk = k + khalf;
    v2f a;
    if (!AVG || kk < 64) {
      a = *(const v2f*)(X0 + ((size_t)mrow << 6) + kk);
    } else {
      size_t off = ((size_t)mrow << 6) + (kk - 64);
      v2f xa = *(const v2f*)(X1a + off);
      v2f xb = *(const v2f*)(X1b + off);
      a.x = 0.5f * (xa.x + xb.x);
      a.y = 0.5f * (xa.y + xb.y);
    }
    // B[kk, n] = W[n, kk]  (out = X @ W^T)
    v2f b0, b1, b2, b3;
    b0.x = sW[(0 * 16 + nl) * K + kk]; b0.y = sW[(0 * 16 + nl) * K + kk + 1];
    b1.x = sW[(1 * 16 + nl) * K + kk]; b1.y = sW[(1 * 16 + nl) * K + kk + 1];
    b2.x = sW[(2 * 16 + nl) * K + kk]; b2.y = sW[(2 * 16 + nl) * K + kk + 1];
    b3.x = sW[(3 * 16 + nl) * K + kk]; b3.y = sW[(3 * 16 + nl) * K + kk + 1];
    acc0 = __builtin_amdgcn_wmma_f32_16x16x4_f32(false, a, false, b0, (short)0, acc0, false, false);
    acc1 = __builtin_amdgcn_wmma_f32_16x16x4_f32(false, a, false, b1, (short)0, acc1, false, false);
    acc2 = __builtin_amdgcn_wmma_f32_16x16x4_f32(false, a, false, b2, (short)0, acc2, false, false);
    acc3 = __builtin_amdgcn_wmma_f32_16x16x4_f32(false, a, false, b3, (short)0, acc3, false, false);
  }

  int rbase = m0 + ((lane >> 4) ? 8 : 0);
  v8f accs[4] = {acc0, acc1, acc2, acc3};
#pragma unroll
  for (int t = 0; t < 4; ++t) {
    int col = t * 16 + nl;
#pragma unroll
    for (int v = 0; v < 8; ++v) {
      float val = accs[t][v];
      if (TANH) val = tanhf(val);
      out[((size_t)(rbase + v) << 6) + col] = val;
    }
  }
}

// column sums (for mean): c[d] += sum over strided rows
__global__ void k_colsum(const float* __restrict__ X, float* __restrict__ c, int Mrows) {
  int d = threadIdx.x;  // 64 threads
  float s = 0.f;
  for (int r = blockIdx.x; r < Mrows; r += gridDim.x) s += X[((size_t)r << 6) + d];
  unsafeAtomicAdd(&c[d], s);
}

// vbuf[0..63]=fk_W@(c1/M), vbuf[64..127]=fk_W@(c2/M), vbuf[128]=fk_b
__global__ void k_makev(const float* __restrict__ fkW, const float* __restrict__ csum,
                        const float* __restrict__ fkb, float* __restrict__ vbuf, float invM) {
  int t = threadIdx.x;  // 128 threads
  int which = t >> 6, r = t & 63;
  const float* c = csum + which * 64;
  float s = 0.f;
  for (int k = 0; k < 64; ++k) s += fkW[r * 64 + k] * (c[k] * invM);
  vbuf[which * 64 + r] = s;
  if (t == 0) vbuf[128] = fkb[0];
}

// logits[(rowOffset+i)*6 + col0 .. col0+ncols-1] = dot(X[i], v) + b
__global__ void k_bilin(const float* __restrict__ X, const float* __restrict__ v,
                        const float* __restrict__ bptr, float* __restrict__ logits,
                        int rowOffset, int col0, int ncols) {
  int i = blockIdx.x * blockDim.x + threadIdx.x;
  if (i >= NUV) return;
  const float4* x4 = (const float4*)(X + ((size_t)i << 6));
  const float4* v4 = (const float4*)v;
  float s = 0.f;
#pragma unroll
  for (int k = 0; k < 16; ++k) {
    float4 a = x4[k], b = v4[k];
    s += a.x * b.x + a.y * b.y + a.z * b.z + a.w * b.w;
  }
  s += bptr[0];
  size_t base = (size_t)(rowOffset + i) * 6 + col0;
  for (int c = 0; c < ncols; ++c) logits[base + c] = s;
}

// Frobenius norm^2 reduction (wave shuffle + one atomic per wave)
__global__ void k_sumsq(const float* __restrict__ X, float* __restrict__ acc, size_t n) {
  size_t stride = (size_t)gridDim.x * blockDim.x;
  float s = 0.f;
  for (size_t i = (size_t)blockIdx.x * blockDim.x + threadIdx.x; i < n; i += stride) {
    float v = X[i];
    s += v * v;
  }
  for (int o = 16; o; o >>= 1) s += __shfl_xor(s, o, 32);
  if ((threadIdx.x & 31) == 0) unsafeAtomicAdd(acc, s);
}

// acc += concat(users * rsqrt(||users||^2), items_next)
__global__ void k_accum(float* __restrict__ acc, const float* __restrict__ users,
                        const float* __restrict__ E1, const float* __restrict__ nrm,
                        size_t nUser, size_t nTot) {
  size_t i = (size_t)blockIdx.x * blockDim.x + threadIdx.x;
  if (i >= nTot) return;
  if (i < nUser) acc[i] += users[i] * rsqrtf(nrm[0]);
  else           acc[i] += E1[i];
}

__global__ void k_final(float* __restrict__ acc, size_t n) {
  size_t i = (size_t)blockIdx.x * blockDim.x + threadIdx.x;
  if (i < n) acc[i] *= 0.25f;
}

// ---------------------------------------------------------------------------
extern "C" void kernel_launch(void* const* d_in, const int* in_sizes, int n_in,
                              void* d_out, int out_size, void* d_ws, size_t ws_size,
                              hipStream_t stream) {
  const float* u1   = (const float*)d_in[0];
  const float* u2   = (const float*)d_in[1];
  const float* item = (const float*)d_in[2];
  const float* Wi   = (const float*)d_in[3];
  const float* Wc   = (const float*)d_in[4];
  const float* Ws   = (const float*)d_in[5];
  const float* fkW  = (const float*)d_in[6];
  const float* fkb  = (const float*)d_in[7];
  const float* Av   = (const float*)d_in[8];
  const float* A2v  = (const float*)d_in[9];
  const float* Sv   = (const float*)d_in[10];
  const int* Ar  = (const int*)d_in[11];
  const int* Ac  = (const int*)d_in[12];
  const int* A2r = (const int*)d_in[13];
  const int* A2c = (const int*)d_in[14];
  const int* Sr  = (const int*)d_in[15];
  const int* Sc  = (const int*)d_in[16];
  const int* sh1 = (const int*)d_in[17];
  const int* sh2 = (const int*)d_in[18];

  const size_t ND = (size_t)NV * DV;   // 16,000,000
  const size_t UD = (size_t)NUV * DV;  //  6,400,000

  float* w = (float*)d_ws;
  float* E0   = w; w += ND;
  float* E1   = w; w += ND;
  float* soc1 = w; w += UD;
  float* soc2 = w; w += UD;
  float* n1a  = w; w += UD;
  float* n1b  = w; w += UD;
  float* n2a  = w; w += UD;
  float* n2b  = w; w += UD;
  float* tmp  = w; w += UD;
  float* csum = w; w += 128;
  float* vbuf = w; w += 132;
  float* nrm  = w; w += 4;

  float* acc    = (float*)d_out;      // ND floats: sum of embs -> mean
  float* logits = acc + ND;           // 2*NU x 6 row-major

  const int T = 256;
  const int gemmBlocks = (NUV / 16 + 7) / 8;  // 6250 stripes, 8 waves/block

  k_init<<<(unsigned)((ND + T - 1) / T), T, 0, stream>>>(u1, u2, item, E0, acc, UD, ND);
  k_gather<<<(unsigned)((UD + T - 1) / T), T, 0, stream>>>(u1, u2, sh1, sh2, n1a, n2a, UD);

  for (int l = 0; l < 3; ++l) {
    const int*   ar = l ? A2r : Ar;
    const int*   ac = l ? A2c : Ac;
    const float* av = l ? A2v : Av;

    // agg = spmm(A, all_emb); then u_next = tanh(agg[:NU] @ Wi^T) in place
    hipMemsetAsync(E1, 0, ND * sizeof(float), stream);
    k_spmm<<<(unsigned)(((long long)NNZ_A * 16 + T - 1) / T), T, 0, stream>>>(ar, ac, av, E0, E1, NNZ_A);
    k_gemm_wmma<64, false, true><<<gemmBlocks, 256, 0, stream>>>(E1, E1, E1, Wi, E1, NUV);

    if (l == 0) {
      // loop-invariant social embeddings + positive logits (cols 0..2)
      hipMemsetAsync(tmp, 0, UD * sizeof(float), stream);
      k_spmm<<<(unsigned)(((long long)NNZ_S * 16 + T - 1) / T), T, 0, stream>>>(Sr, Sc, Sv, u1, tmp, NNZ_S);
      k_gemm_wmma<64, false, true><<<gemmBlocks, 256, 0, stream>>>(tmp, tmp, tmp, Wc, soc1, NUV);
      hipMemsetAsync(tmp, 0, UD * sizeof(float), stream);
      k_spmm<<<(unsigned)(((long long)NNZ_S * 16 + T - 1) / T), T, 0, stream>>>(Sr, Sc, Sv, u2, tmp, NNZ_S);
      k_gemm_wmma<64, false, true><<<gemmBlocks, 256, 0, stream>>>(tmp, tmp, tmp, Wc, soc2, NUV);

      hipMemsetAsync(csum, 0, 128 * sizeof(float), stream);
      k_colsum<<<256, 64, 0, stream>>>(soc1, csum, NUV);
      k_colsum<<<256, 64, 0, stream>>>(soc2, csum + 64, NUV);
      k_makev<<<1, 128, 0, stream>>>(fkW, csum, fkb, vbuf, 1.0f / NUV);

      k_bilin<<<(NUV + T - 1) / T, T, 0, stream>>>(soc2, vbuf, vbuf + 128, logits, 0,   0, 3);
      k_bilin<<<(NUV + T - 1) / T, T, 0, stream>>>(soc1, vbuf + 64, vbuf + 128, logits, NUV, 0, 3);
    }

    // negative streams: n = tanh(spmm(S, n) @ Wc^T)
    hipMemsetAsync(n1b, 0, UD * sizeof(float), stream);
    k_spmm<<<(unsigned)(((long long)NNZ_S * 16 + T - 1) / T), T, 0, stream>>>(Sr, Sc, Sv, n1a, n1b, NNZ_S);
    k_gemm_wmma<64, false, true><<<gemmBlocks, 256, 0, stream>>>(n1b, n1b, n1b, Wc, n1b, NUV);
    { float* t2 = n1a; n1a = n1b; n1b = t2; }

    hipMemsetAsync(n2b, 0, UD * sizeof(float), stream);
    k_spmm<<<(unsigned)(((long long)NNZ_S * 16 + T - 1) / T), T, 0, stream>>>(Sr, Sc, Sv, n2a, n2b, NNZ_S);
    k_gemm_wmma<64, false, true><<<gemmBlocks, 256, 0, stream>>>(n2b, n2b, n2b, Wc, n2b, NUV);
    { float* t2 = n2a; n2a = n2b; n2b = t2; }

    // negative logits (col 3+l)
    k_bilin<<<(NUV + T - 1) / T, T, 0, stream>>>(n2a, vbuf, vbuf + 128, logits, 0,   3 + l, 1);
    k_bilin<<<(NUV + T - 1) / T, T, 0, stream>>>(n1a, vbuf + 64, vbuf + 128, logits, NUV, 3 + l, 1);

    // users = concat(u_next, 0.5*(soc1+soc2)) @ Ws^T ; Frobenius normalize; accumulate
    k_gemm_wmma<128, true, false><<<gemmBlocks, 256, 0, stream>>>(E1, soc1, soc2, Ws, tmp, NUV);
    hipMemsetAsync(nrm, 0, sizeof(float), stream);
    k_sumsq<<<2048, T, 0, stream>>>(tmp, nrm, UD);
    k_accum<<<(unsigned)((ND + T - 1) / T), T, 0, stream>>>(acc, tmp, E1, nrm, UD, ND);

    { float* t2 = E0; E0 = E1; E1 = t2; }  // all_emb = concat(u_next, items_next)
  }

  k_final<<<(unsigned)((ND + T - 1) / T), T, 0, stream>>>(acc, ND);
}